// Student_Net_80874234183991
// MI455X (gfx1250) — compile-verified
//
#include <hip/hip_runtime.h>

typedef __attribute__((ext_vector_type(2))) float v2f;
typedef __attribute__((ext_vector_type(8))) float v8f;

#define N_NODES_C 50000
#define N_EDGES_C 800000
#define IN_DIM_C 300
#define FEAT_DIM_C 200
#define OUT_DIM_C 64
#define LD1_C 208            // padded row stride for 200-wide activations (13 tiles)
#define BN_EPS_C 1e-3f
#define ALPHA_C 0.2f

__device__ __forceinline__ v8f wmma_f32(v2f a, v2f b, v8f c) {
  // V_WMMA_F32_16X16X4_F32 : D = A(16x4) * B(4x16) + C(16x16), wave32
  return __builtin_amdgcn_wmma_f32_16x16x4_f32(false, a, false, b, (short)0, c,
                                               false, false);
}

// Pack W[K,N] (row major) into WMMA B-fragment layout, zero-padded to tilesN*16:
//   Bp[((tn*ksteps + ks)*32 + lane)*2 + e] = W[(4*ks + 2*(lane>>4) + e), tn*16 + (lane&15)]
__global__ void kpackB(const float* __restrict__ W, float* __restrict__ Bp,
                       int K, int N, int tilesN) {
  const int ksteps = K >> 2;
  const int idx = blockIdx.x * blockDim.x + threadIdx.x;
  if (idx >= tilesN * ksteps * 32) return;
  const int lane = idx & 31;
  const int ks = (idx >> 5) % ksteps;
  const int tn = (idx >> 5) / ksteps;
  const int col = (tn << 4) + (lane & 15);
  const int k = (ks << 2) + ((lane >> 4) << 1);
  const bool ok = col < N;
  Bp[(size_t)idx * 2 + 0] = ok ? W[(size_t)k * N + col] : 0.f;
  Bp[(size_t)idx * 2 + 1] = ok ? W[(size_t)(k + 1) * N + col] : 0.f;
}

// C[M, tilesN*16] = A[M,K] * Bpacked ; all loads/stores unguarded.
// M multiple of 16, K multiple of 4, ldc == tilesN*16.
__global__ void gemm_nn_packed(const float* __restrict__ A, int lda,
                               const float* __restrict__ Bp,
                               float* __restrict__ C, int ldc,
                               int M, int K, int tilesN) {
  const int lane = threadIdx.x & 31;
  const int half = lane >> 4;
  const int l = lane & 15;
  const int tilesM = M >> 4;
  const int wave = (blockIdx.x * blockDim.x + threadIdx.x) >> 5;
  if (wave >= tilesM * tilesN) return;
  const int tm = wave / tilesN, tn = wave % tilesN;
  const int m0 = tm << 4;
  const int col = (tn << 4) + l;
  const int ksteps = K >> 2;
  const float* Arow = A + (size_t)(m0 + l) * lda + (half << 1);
  const v2f* bp = (const v2f*)(Bp + (size_t)tn * ksteps * 64) + lane;
  v8f acc = {0.f, 0.f, 0.f, 0.f, 0.f, 0.f, 0.f, 0.f};
#pragma unroll 4
  for (int ks = 0; ks < ksteps; ++ks) {
    v2f a;
    a.x = Arow[0];
    a.y = Arow[1];
    v2f b = *bp;
    acc = wmma_f32(a, b, acc);
    Arow += 4;
    bp += 32;
  }
  float* Cp = C + (size_t)(m0 + (half << 3)) * ldc + col;
#pragma unroll
  for (int v = 0; v < 8; ++v) Cp[(size_t)v * ldc] = acc[v];
}

// G[tiles*16, tiles*16] += H^T H over a K slice (split-K, atomics).
// H is ldh-strided with zero-padded columns up to tiles*16 -> no guards.
__global__ void gram_atomic(const float* __restrict__ H, int ldh,
                            float* __restrict__ G, int ldg,
                            int tiles, int Nrows, int kchunk) {
  const int lane = threadIdx.x & 31;
  const int half = lane >> 4;
  const int l = lane & 15;
  const int wave = (blockIdx.x * blockDim.x + threadIdx.x) >> 5;
  if (wave >= tiles * tiles) return;
  const int tm = wave / tiles, tn = wave % tiles;
  const int ca = (tm << 4) + l;
  const int cb = (tn << 4) + l;
  const int kbeg = blockIdx.y * kchunk;
  int kend = kbeg + kchunk;
  if (kend > Nrows) kend = Nrows;
  v8f acc = {0.f, 0.f, 0.f, 0.f, 0.f, 0.f, 0.f, 0.f};
  const float* H0 = H + (size_t)(kbeg + (half << 1)) * ldh;
#pragma unroll 2
  for (int k0 = kbeg; k0 < kend; k0 += 4) {
    v2f a, b;
    a.x = H0[ca];
    a.y = H0[ldh + ca];
    b.x = H0[cb];
    b.y = H0[ldh + cb];
    acc = wmma_f32(a, b, acc);
    H0 += 4 * (size_t)ldh;
  }
  float* Gp = G + (size_t)((tm << 4) + (half << 3)) * ldg + cb;
#pragma unroll
  for (int v = 0; v < 8; ++v) atomicAdd(&Gp[(size_t)v * ldg], acc[v]);
}

// ---------------- utility kernels ------------------------------------------
__global__ void kzero(float* p, long long n) {
  long long i = (long long)blockIdx.x * blockDim.x + threadIdx.x;
  if (i < n) p[i] = 0.f;
}

__global__ void kdeg(const int* __restrict__ dst, float* __restrict__ deg, int nE) {
  int e = blockIdx.x * blockDim.x + threadIdx.x;
  if (e < nE) atomicAdd(&deg[dst[e]], 1.0f);
}

// agg[dst] += P[src]; D multiple of 4; ld-strided rows
__global__ void kscatter(const float* __restrict__ P, const int* __restrict__ src,
                         const int* __restrict__ dst, float* __restrict__ agg,
                         int nE, int D, int ld) {
  const int chunks = D >> 2;
  long long t = (long long)blockIdx.x * blockDim.x + threadIdx.x;
  if (t >= (long long)nE * chunks) return;
  int e = (int)(t / chunks);
  int c = (int)(t % chunks) << 2;
  const float* p = P + (size_t)src[e] * ld + c;
  float* a = agg + (size_t)dst[e] * ld + c;
  atomicAdd(a + 0, p[0]);
  atomicAdd(a + 1, p[1]);
  atomicAdd(a + 2, p[2]);
  atomicAdd(a + 3, p[3]);
}

// O = S + agg/max(deg,1) + bias  (O may alias S); only logical columns j<D
__global__ void kcombine(const float* S, const float* agg,
                         const float* __restrict__ deg,
                         const float* __restrict__ bias, float* O,
                         int Nn, int D, int ld) {
  long long idx = (long long)blockIdx.x * blockDim.x + threadIdx.x;
  if (idx >= (long long)Nn * D) return;
  int i = (int)(idx / D), j = (int)(idx % D);
  size_t a = (size_t)i * ld + j;
  O[a] = S[a] + agg[a] / fmaxf(deg[i], 1.0f) + bias[j];
}

__global__ void kcolstats(const float* __restrict__ H, float* __restrict__ mean,
                          float* __restrict__ var, int Nrows, int ld) {
  const int j = blockIdx.x;
  __shared__ float s1[256], s2[256];
  float a = 0.f, b = 0.f;
  for (int i = threadIdx.x; i < Nrows; i += blockDim.x) {
    float x = H[(size_t)i * ld + j];
    a += x;
    b += x * x;
  }
  s1[threadIdx.x] = a;
  s2[threadIdx.x] = b;
  __syncthreads();
  for (int off = blockDim.x >> 1; off > 0; off >>= 1) {
    if (threadIdx.x < off) {
      s1[threadIdx.x] += s1[threadIdx.x + off];
      s2[threadIdx.x] += s2[threadIdx.x + off];
    }
    __syncthreads();
  }
  if (threadIdx.x == 0) {
    float mu = s1[0] / (float)Nrows;
    mean[j] = mu;
    var[j] = fmaxf(s2[0] / (float)Nrows - mu * mu, 0.f);
  }
}

__global__ void kbnleaky(float* __restrict__ H, const float* __restrict__ mean,
                         const float* __restrict__ var,
                         const float* __restrict__ gamma,
                         const float* __restrict__ beta, int Nn, int D, int ld) {
  long long idx = (long long)blockIdx.x * blockDim.x + threadIdx.x;
  if (idx >= (long long)Nn * D) return;
  int i = (int)(idx / D), j = (int)(idx % D);
  size_t a = (size_t)i * ld + j;
  float x = H[a];
  x = (x - mean[j]) * rsqrtf(var[j] + BN_EPS_C) * gamma[j] + beta[j];
  H[a] = x >= 0.f ? x : ALPHA_C * x;
}

// l2-normalize each logical row (length D) of ldg-strided G; one block per row
__global__ void kl2rows(float* __restrict__ G, int D, int ldg) {
  float* row = G + (size_t)blockIdx.x * ldg;
  __shared__ float red[256];
  float s = 0.f;
  for (int j = threadIdx.x; j < D; j += blockDim.x) {
    float x = row[j];
    s += x * x;
  }
  red[threadIdx.x] = s;
  __syncthreads();
  for (int off = blockDim.x >> 1; off; off >>= 1) {
    if (threadIdx.x < off) red[threadIdx.x] += red[threadIdx.x + off];
    __syncthreads();
  }
  float inv = rsqrtf(fmaxf(red[0], 1e-12f));
  for (int j = threadIdx.x; j < D; j += blockDim.x) row[j] *= inv;
}

// power iteration + the reference's [d,1] "row l2norm" (elementwise sign)
__global__ void kpower(const float* __restrict__ G, int ldg,
                       const float* __restrict__ rin, float* __restrict__ rout,
                       float* __restrict__ scale, int D) {
  __shared__ float sr[256];
  __shared__ float red[256];
  const int t = threadIdx.x;
  if (t < D) sr[t] = rin[t];
  __syncthreads();
  const float invkd = 1.0f / (4.0f * (float)D);
  for (int it = 0; it < 4; ++it) {
    float s = 0.f;
    if (t < D) {
      const float* row = G + (size_t)t * ldg;
      for (int j = 0; j < D; ++j) s += row[j] * sr[j];
    }
    __syncthreads();
    if (t < D) sr[t] = s * invkd;
    __syncthreads();
  }
  float rv = 0.f;
  if (t < D) {
    float x = sr[t];
    rv = x * rsqrtf(fmaxf(x * x, 1e-12f));
  }
  red[t] = rv * rv;
  __syncthreads();
  for (int off = blockDim.x >> 1; off; off >>= 1) {
    if (t < off) red[t] += red[t + off];
    __syncthreads();
  }
  if (t < D) rout[t] = rv;
  if (t == 0) scale[0] = 1.0f / ((float)D * (red[0] + 1e-6f));
}

// H[row] -= (H[row].r)*scale * r ; then l2-normalize the row. One block/row.
__global__ void ksfa(float* __restrict__ H, int ld, const float* __restrict__ r,
                     const float* __restrict__ scale, int D) {
  float* Hr = H + (size_t)blockIdx.x * ld;
  __shared__ float red[64];
  const int t = threadIdx.x;
  float s = 0.f;
  for (int j = t; j < D; j += 64) s += Hr[j] * r[j];
  red[t] = s;
  __syncthreads();
  for (int off = 32; off; off >>= 1) {
    if (t < off) red[t] += red[t + off];
    __syncthreads();
  }
  const float c = red[0] * scale[0];
  __syncthreads();
  float s2 = 0.f;
  for (int j = t; j < D; j += 64) {
    float x = Hr[j] - c * r[j];
    Hr[j] = x;
    s2 += x * x;
  }
  red[t] = s2;
  __syncthreads();
  for (int off = 32; off; off >>= 1) {
    if (t < off) red[t] += red[t + off];
    __syncthreads();
  }
  const float inv = rsqrtf(fmaxf(red[0], 1e-12f));
  for (int j = t; j < D; j += 64) Hr[j] *= inv;
}

static inline int cdiv(long long a, int b) { return (int)((a + b - 1) / b); }

extern "C" void kernel_launch(void* const* d_in, const int* in_sizes, int n_in,
                              void* d_out, int out_size, void* d_ws, size_t ws_size,
                              hipStream_t stream) {
  const float* feats   = (const float*)d_in[0];
  const float* Wself1  = (const float*)d_in[1];
  const float* Wneigh1 = (const float*)d_in[2];
  const float* b1      = (const float*)d_in[3];
  const float* gamma1  = (const float*)d_in[4];
  const float* beta1   = (const float*)d_in[5];
  const float* Wself2  = (const float*)d_in[6];
  const float* Wneigh2 = (const float*)d_in[7];
  const float* b2      = (const float*)d_in[8];
  const float* gamma2  = (const float*)d_in[9];
  const float* beta2   = (const float*)d_in[10];
  const float* r1      = (const float*)d_in[11];
  const float* r2      = (const float*)d_in[12];
  const int* esrc      = (const int*)d_in[13];
  const int* edst      = (const int*)d_in[14];
  float* out = (float*)d_out;

  // ---- workspace layout (floats), ~126 MB ----
  float* w = (float*)d_ws;
  const size_t NBP = (size_t)N_NODES_C * LD1_C;  // 10,400,000 (padded stride 208)
  float* S1    = w;                  // [50000,208] self-proj -> H1 in place
  float* P1    = w + NBP;            // [50000,208] neigh-proj; layer2: P2 + agg2
  float* agg1  = w + 2 * NBP;        // [50000,208]; layer2: S2
  float* degb  = w + 3 * NBP;        // [50000]
  float* G     = degb + N_NODES_C;   // [208,208]
  float* rbuf  = G + LD1_C * LD1_C;
  float* meanb = rbuf + 256;
  float* varb  = meanb + 256;
  float* scal  = varb + 256;
  float* BpS1  = scal + 16;          // 13*75*64 = 62400
  float* BpN1  = BpS1 + 13 * 75 * 64;
  float* BpS2  = BpN1 + 13 * 75 * 64;  // 4*50*64 = 12800
  float* BpN2  = BpS2 + 4 * 50 * 64;
  float* agg2  = P1 + NBP / 2;       // [50000,64] upper half of P1 buffer
  float* S2    = agg1;
  float* P2    = P1;

  const int B = 256;
  const int T1 = 13, T2 = 4;  // N tiles for 208 / 64

  // ---- pack weights into WMMA fragment layout (zero-padded) ----
  kpackB<<<cdiv(T1 * 75 * 32, B), B, 0, stream>>>(Wself1, BpS1, IN_DIM_C, FEAT_DIM_C, T1);
  kpackB<<<cdiv(T1 * 75 * 32, B), B, 0, stream>>>(Wneigh1, BpN1, IN_DIM_C, FEAT_DIM_C, T1);
  kpackB<<<cdiv(T2 * 50 * 32, B), B, 0, stream>>>(Wself2, BpS2, FEAT_DIM_C, OUT_DIM_C, T2);
  kpackB<<<cdiv(T2 * 50 * 32, B), B, 0, stream>>>(Wneigh2, BpN2, FEAT_DIM_C, OUT_DIM_C, T2);

  // ---------------- layer 1 ----------------
  {
    int waves = (N_NODES_C >> 4) * T1;  // 40625
    int blk = cdiv((long long)waves * 32, B);
    gemm_nn_packed<<<blk, B, 0, stream>>>(feats, IN_DIM_C, BpS1, S1, LD1_C,
                                          N_NODES_C, IN_DIM_C, T1);
    gemm_nn_packed<<<blk, B, 0, stream>>>(feats, IN_DIM_C, BpN1, P1, LD1_C,
                                          N_NODES_C, IN_DIM_C, T1);
  }
  kzero<<<cdiv(N_NODES_C, B), B, 0, stream>>>(degb, N_NODES_C);
  kzero<<<cdiv((long long)NBP, B), B, 0, stream>>>(agg1, (long long)NBP);
  kdeg<<<cdiv(N_EDGES_C, B), B, 0, stream>>>(edst, degb, N_EDGES_C);
  kscatter<<<cdiv((long long)N_EDGES_C * (FEAT_DIM_C / 4), B), B, 0, stream>>>(
      P1, esrc, edst, agg1, N_EDGES_C, FEAT_DIM_C, LD1_C);
  const long long NL1 = (long long)N_NODES_C * FEAT_DIM_C;  // logical elems
  kcombine<<<cdiv(NL1, B), B, 0, stream>>>(S1, agg1, degb, b1, S1, N_NODES_C,
                                           FEAT_DIM_C, LD1_C);
  kcolstats<<<FEAT_DIM_C, B, 0, stream>>>(S1, meanb, varb, N_NODES_C, LD1_C);
  kbnleaky<<<cdiv(NL1, B), B, 0, stream>>>(S1, meanb, varb, gamma1, beta1,
                                           N_NODES_C, FEAT_DIM_C, LD1_C);
  // SFA 1
  kzero<<<cdiv(LD1_C * LD1_C, B), B, 0, stream>>>(G, LD1_C * LD1_C);
  {
    dim3 grid(cdiv((long long)(T1 * T1) * 32, B), 25);  // split-K: 25 x 2000 rows
    gram_atomic<<<grid, B, 0, stream>>>(S1, LD1_C, G, LD1_C, T1, N_NODES_C, 2000);
  }
  kl2rows<<<FEAT_DIM_C, B, 0, stream>>>(G, FEAT_DIM_C, LD1_C);
  kpower<<<1, B, 0, stream>>>(G, LD1_C, r1, rbuf, scal, FEAT_DIM_C);
  ksfa<<<N_NODES_C, 64, 0, stream>>>(S1, LD1_C, rbuf, scal, FEAT_DIM_C);

  // ---------------- layer 2 ----------------
  {
    int waves = (N_NODES_C >> 4) * T2;  // 12500
    int blk = cdiv((long long)waves * 32, B);
    gemm_nn_packed<<<blk, B, 0, stream>>>(S1, LD1_C, BpS2, S2, OUT_DIM_C,
                                          N_NODES_C, FEAT_DIM_C, T2);
    gemm_nn_packed<<<blk, B, 0, stream>>>(S1, LD1_C, BpN2, P2, OUT_DIM_C,
                                          N_NODES_C, FEAT_DIM_C, T2);
  }
  const long long NO = (long long)N_NODES_C * OUT_DIM_C;  // 3,200,000
  kzero<<<cdiv(NO, B), B, 0, stream>>>(agg2, NO);
  kscatter<<<cdiv((long long)N_EDGES_C * (OUT_DIM_C / 4), B), B, 0, stream>>>(
      P2, esrc, edst, agg2, N_EDGES_C, OUT_DIM_C, OUT_DIM_C);
  kcombine<<<cdiv(NO, B), B, 0, stream>>>(S2, agg2, degb, b2, out, N_NODES_C,
                                          OUT_DIM_C, OUT_DIM_C);
  kcolstats<<<OUT_DIM_C, B, 0, stream>>>(out, meanb, varb, N_NODES_C, OUT_DIM_C);
  kbnleaky<<<cdiv(NO, B), B, 0, stream>>>(out, meanb, varb, gamma2, beta2,
                                          N_NODES_C, OUT_DIM_C, OUT_DIM_C);
  // SFA 2
  kzero<<<cdiv(OUT_DIM_C * OUT_DIM_C, B), B, 0, stream>>>(G, OUT_DIM_C * OUT_DIM_C);
  {
    dim3 grid(cdiv((long long)(T2 * T2) * 32, B), 25);
    gram_atomic<<<grid, B, 0, stream>>>(out, OUT_DIM_C, G, OUT_DIM_C, T2,
                                        N_NODES_C, 2000);
  }
  kl2rows<<<OUT_DIM_C, B, 0, stream>>>(G, OUT_DIM_C, OUT_DIM_C);
  kpower<<<1, B, 0, stream>>>(G, OUT_DIM_C, r2, rbuf, scal, OUT_DIM_C);
  ksfa<<<N_NODES_C, 64, 0, stream>>>(out, OUT_DIM_C, rbuf, scal, OUT_DIM_C);
}